// ProposalTarget_20169166422191
// MI455X (gfx1250) — compile-verified
//
#include <hip/hip_runtime.h>
#include <hip/hip_bf16.h>

// ProposalTarget for MI455X (gfx1250, wave32).
// Stage 1: IoU + argmax (GT tile staged via TDM tensor_load_to_lds).
// Stage 2: per-image 16K-key bitonic sort in 128KB LDS -> exact stable top-k.
// Stage 3: gather + delta-encode + one-hot expansion into the 3 flat outputs.

#define B_IMG   32
#define N_ROI   12000
#define G_GT    200
#define BG_CLS  20
#define NUM_POS 64
#define ROI_BS  256
#define SORT_N  16384   // next pow2 >= N_ROI; 16384 * 8B = 128 KB LDS (fits 320KB WGP LDS)

typedef unsigned int u32x4 __attribute__((ext_vector_type(4)));
typedef int          i32x8 __attribute__((ext_vector_type(8)));
typedef int          i32x4 __attribute__((ext_vector_type(4)));

// ---------------------------------------------------------------------------
// Stage 1: per-ROI max IoU + best GT. One block = 256 ROIs of one image.
// GT boxes (800 f32 = 3.2KB) are DMA'd into LDS by the Tensor Data Mover.
// ---------------------------------------------------------------------------
__global__ __launch_bounds__(256) void pt_iou_argmax(
    const float* __restrict__ rois,      // [B, N, 4]
    const int*   __restrict__ gtcls,     // [B, G]
    const float* __restrict__ gtb,       // [B, G, 4]
    float*       __restrict__ maxIou,    // [B, N]
    int*         __restrict__ bestGt)    // [B, N]
{
  __shared__ float sGT[G_GT * 4];
  __shared__ int   sValid[G_GT];
  const int b = blockIdx.y;

  // Wave 0 issues a TDM load of the GT-box tile into LDS (1 row x 800 f32).
  if (threadIdx.x < 32) {
    const float* src = gtb + (size_t)b * G_GT * 4;
    unsigned long long ga = (unsigned long long)(size_t)src;
    // Flat LDS pointer: low 32 bits == LDS byte offset (aperture in high bits).
    unsigned lds = (unsigned)(unsigned long long)(size_t)(&sGT[0]);
    // D# group 0: count=1 | lds_addr | global_addr[56:0] | type=2
    u32x4 g0;
    g0.x = 1u;                                   // count=1, no gather
    g0.y = lds;                                  // lds_addr
    g0.z = (unsigned)(ga & 0xFFFFFFFFull);       // global_addr[31:0]
    g0.w = (unsigned)((ga >> 32) & 0x01FFFFFFull) | (2u << 30); // addr[56:32], type=2
    // D# group 1: data_size=4B, tensor_dim0=800, tensor_dim1=1,
    //             tile_dim0=800, tile_dim1=1, dim0_stride=800
    i32x8 g1;
    g1[0] = (int)(2u << 16);      // workgroup_mask=0, data_size=2 (4 bytes)
    g1[1] = (int)(800u << 16);    // tensor_dim0 low16 (bits 63:48)
    g1[2] = (int)(1u << 16);      // tensor_dim0 hi=0, tensor_dim1 low16 = 1
    g1[3] = (int)(800u << 16);    // tensor_dim1 hi=0, tile_dim0 = 800
    g1[4] = 1;                    // tile_dim1 = 1, tile_dim2 = 0
    g1[5] = 800;                  // tensor_dim0_stride low32
    g1[6] = 0;                    // stride hi, dim1_stride low
    g1[7] = 0;
    i32x4 z4 = {0, 0, 0, 0};
    i32x8 z8 = {0, 0, 0, 0, 0, 0, 0, 0};
    // 6-arg form (clang-23 / therock-10.0 headers): g0, g1, g2, g3, g4, cpol
    __builtin_amdgcn_tensor_load_to_lds(g0, g1, z4, z4, z8, 0);
    __builtin_amdgcn_s_wait_tensorcnt((short)0);   // wave-local TENSORcnt drain
  }
  for (int i = threadIdx.x; i < G_GT; i += blockDim.x)
    sValid[i] = (gtcls[b * G_GT + i] < BG_CLS) ? 1 : 0;
  __syncthreads();   // other waves wait for wave 0's TDM completion here

  const int n = blockIdx.x * blockDim.x + threadIdx.x;
  if (n >= N_ROI) return;

  const float* rp = rois + ((size_t)b * N_ROI + n) * 4;
  __builtin_prefetch(rp, 0, 3);                    // global_prefetch_b8
  const float4 r = *(const float4*)rp;             // (y1,x1,y2,x2)
  const float ar = (r.z - r.x) * (r.w - r.y);

  float best = -2.0f;
  int   bi   = 0;
  for (int g = 0; g < G_GT; ++g) {
    const float gy1 = sGT[g * 4 + 0], gx1 = sGT[g * 4 + 1];
    const float gy2 = sGT[g * 4 + 2], gx2 = sGT[g * 4 + 3];
    const float y1 = fmaxf(r.x, gy1), x1 = fmaxf(r.y, gx1);
    const float y2 = fminf(r.z, gy2), x2 = fminf(r.w, gx2);
    const float inter = fmaxf(y2 - y1, 0.0f) * fmaxf(x2 - x1, 0.0f);
    const float ag = (gy2 - gy1) * (gx2 - gx1);
    float iou = inter / (ar + ag - inter + 1e-8f);
    iou = sValid[g] ? iou : -1.0f;
    if (iou > best) { best = iou; bi = g; }        // first-max == jnp.argmax
  }
  maxIou[(size_t)b * N_ROI + n] = best;
  bestGt[(size_t)b * N_ROI + n] = bi;
}

// ---------------------------------------------------------------------------
// Stage 2: exact stable top-k via full bitonic sort of packed 64-bit keys.
// key = cat(2b: 2=pos,1=neg,0=invalid) << 46 | ord(iou) << 14 | (16383-idx)
// Descending sort => pos desc-IoU asc-idx, then neg, then invalid.
// ---------------------------------------------------------------------------
__global__ __launch_bounds__(1024) void pt_select(
    const float* __restrict__ maxIou,    // [B, N]
    int* __restrict__ selIdx,            // [B, ROI_BS]
    int* __restrict__ selPos)            // [B, ROI_BS]
{
  extern __shared__ unsigned long long keys[];     // SORT_N * 8B = 128 KB
  __shared__ int sFirstNonPos, sFirstNonNeg;
  const int b = blockIdx.x;
  const float* mi = maxIou + (size_t)b * N_ROI;

  for (int i = threadIdx.x; i < SORT_N; i += blockDim.x) {
    unsigned long long k = 0ull;
    if (i < N_ROI) {
      if ((i & 63) == 0) __builtin_prefetch(mi + i + 64, 0, 3);
      const float v = mi[i];
      const unsigned u = __float_as_uint(v);
      const unsigned ord = (u & 0x80000000u) ? ~u : (u | 0x80000000u);
      const unsigned cat = (v >= 0.5f) ? 2u : ((v >= 0.0f) ? 1u : 0u);
      k = ((unsigned long long)cat << 46) | ((unsigned long long)ord << 14) |
          (unsigned long long)(16383 - i);
    }
    keys[i] = k;
  }
  if (threadIdx.x == 0) { sFirstNonPos = SORT_N; sFirstNonNeg = SORT_N; }
  __syncthreads();

  // Bitonic sort, descending.
  for (int k = 2; k <= SORT_N; k <<= 1) {
    for (int j = k >> 1; j > 0; j >>= 1) {
      for (int i = threadIdx.x; i < SORT_N; i += blockDim.x) {
        const int ixj = i ^ j;
        if (ixj > i) {
          const unsigned long long a = keys[i];
          const unsigned long long c = keys[ixj];
          const bool up = ((i & k) == 0);
          if (up ? (a < c) : (a > c)) { keys[i] = c; keys[ixj] = a; }
        }
      }
      __syncthreads();
    }
  }

  // Category boundaries (categories are contiguous after the sort).
  for (int i = threadIdx.x; i < SORT_N; i += blockDim.x) {
    const unsigned cat = (unsigned)(keys[i] >> 46);
    if (cat < 2u) atomicMin(&sFirstNonPos, i);
    if (cat < 1u) atomicMin(&sFirstNonNeg, i);
  }
  __syncthreads();

  const int nPos    = sFirstNonPos;
  const int nPosNeg = sFirstNonNeg;
  const int nNeg    = nPosNeg - nPos;
  const int takePos = (nPos < NUM_POS) ? nPos : NUM_POS;
  const int negAvail = (nNeg < ROI_BS) ? nNeg : ROI_BS;

  for (int s = threadIdx.x; s < ROI_BS; s += blockDim.x) {
    unsigned long long e;
    int isp = 0;
    if (s < takePos) {                       // valid positives first
      e = keys[s]; isp = 1;
    } else {
      const int ns = s - takePos;
      if (ns < negAvail) {                   // then valid negatives
        e = keys[nPos + ns];
      } else {                               // filler (practically unreachable)
        int fi = nPosNeg + (ns - negAvail);
        if (fi > SORT_N - 1) fi = SORT_N - 1;
        e = keys[fi];
      }
    }
    int idx = 16383 - (int)(e & 0x3FFFull);
    if (idx >= N_ROI || idx < 0) idx = 0;    // padding-slot guard
    selIdx[b * ROI_BS + s] = idx;
    selPos[b * ROI_BS + s] = isp;
  }
}

// ---------------------------------------------------------------------------
// Stage 3: gather + encode + one-hot expansion. One thread per output row.
// Writes every element of d_out (no zero-init dependency).
// ---------------------------------------------------------------------------
__global__ __launch_bounds__(ROI_BS) void pt_emit(
    const float* __restrict__ rois,      // [B, N, 4]
    const int*   __restrict__ gtcls,     // [B, G]
    const float* __restrict__ gtb,       // [B, G, 4]
    const int*   __restrict__ bestGt,    // [B, N]
    const int*   __restrict__ selIdx,    // [B, ROI_BS]
    const int*   __restrict__ selPos,    // [B, ROI_BS]
    float*       __restrict__ out)       // rois_sel ++ cls_onehot ++ reg
{
  const int b = blockIdx.x;
  const int s = threadIdx.x;
  const int row = b * ROI_BS + s;

  const int n   = selIdx[row];
  const int isp = selPos[row];
  const int g   = bestGt[(size_t)b * N_ROI + n];
  const int label = isp ? gtcls[b * G_GT + g] : BG_CLS;

  const float4 r  = *(const float4*)(rois + ((size_t)b * N_ROI + n) * 4);
  const float4 gt = *(const float4*)(gtb  + ((size_t)b * G_GT  + g) * 4);

  // Output 0: rois_sel [B,256,4]
  float* o0 = out + (size_t)row * 4;
  o0[0] = r.x; o0[1] = r.y; o0[2] = r.z; o0[3] = r.w;

  // Output 1: cls_onehot [B,256,21]
  float* o1 = out + (size_t)B_IMG * ROI_BS * 4 + (size_t)row * (BG_CLS + 1);
  #pragma unroll
  for (int c = 0; c <= BG_CLS; ++c) o1[c] = (c == label) ? 1.0f : 0.0f;

  // Delta encoding (tx,ty,tw,th) * (10,10,5,5), zeroed for non-positives.
  const float eps = 1e-6f;
  const float rh = r.z - r.x, rw = r.w - r.y;
  const float rcy = r.x + rh * 0.5f, rcx = r.y + rw * 0.5f;
  const float gh = gt.z - gt.x, gw = gt.w - gt.y;
  const float gcy = gt.x + gh * 0.5f, gcx = gt.y + gw * 0.5f;
  const float fp = (float)isp;
  float t[4];
  t[0] = (gcx - rcx) / (rw + eps) * 10.0f * fp;
  t[1] = (gcy - rcy) / (rh + eps) * 10.0f * fp;
  t[2] = logf(fmaxf(gw, eps) / (rw + eps)) * 5.0f * fp;
  t[3] = logf(fmaxf(gh, eps) / (rh + eps)) * 5.0f * fp;

  // Output 2: reg [B,256,160] = mask4 (80) ++ deltas_full (80)
  float* o2 = out + (size_t)B_IMG * ROI_BS * 4 +
              (size_t)B_IMG * ROI_BS * (BG_CLS + 1) + (size_t)row * (8 * BG_CLS);
  #pragma unroll
  for (int c = 0; c < BG_CLS; ++c) {
    const float m = (c == label) ? 1.0f : 0.0f;
    o2[c * 4 + 0] = m; o2[c * 4 + 1] = m; o2[c * 4 + 2] = m; o2[c * 4 + 3] = m;
    o2[4 * BG_CLS + c * 4 + 0] = m * t[0];
    o2[4 * BG_CLS + c * 4 + 1] = m * t[1];
    o2[4 * BG_CLS + c * 4 + 2] = m * t[2];
    o2[4 * BG_CLS + c * 4 + 3] = m * t[3];
  }
}

// ---------------------------------------------------------------------------
extern "C" void kernel_launch(void* const* d_in, const int* in_sizes, int n_in,
                              void* d_out, int out_size, void* d_ws, size_t ws_size,
                              hipStream_t stream) {
  const float* rois  = (const float*)d_in[0];  // [32,12000,4]
  const int*   gtcls = (const int*)  d_in[1];  // [32,200]
  const float* gtb   = (const float*)d_in[2];  // [32,200,4]
  float* out = (float*)d_out;

  // Workspace layout (~3.1 MB)
  float* maxIou = (float*)d_ws;                                  // B*N f32
  int*   bestGt = (int*)(maxIou + (size_t)B_IMG * N_ROI);        // B*N i32
  int*   selIdx = bestGt + (size_t)B_IMG * N_ROI;                // B*256 i32
  int*   selPos = selIdx + (size_t)B_IMG * ROI_BS;               // B*256 i32

  dim3 g1((N_ROI + 255) / 256, B_IMG);
  pt_iou_argmax<<<g1, 256, 0, stream>>>(rois, gtcls, gtb, maxIou, bestGt);

  pt_select<<<B_IMG, 1024, SORT_N * sizeof(unsigned long long), stream>>>(
      maxIou, selIdx, selPos);

  pt_emit<<<B_IMG, ROI_BS, 0, stream>>>(rois, gtcls, gtb, bestGt,
                                        selIdx, selPos, out);
}